// GraphFeature_12996571037964
// MI455X (gfx1250) — compile-verified
//
#include <hip/hip_runtime.h>

// GraphFeature (DGCNN knn + edge features) for MI455X / gfx1250, wave32.
// x: (8, 64, 4096) f32.  out: (8, 128, 4096, 20) f32.
//
// Kernel A: pairwise -squared-distance via V_WMMA_F32_16X16X4_F32 (K=3 padded
//           to 4), chunked through LDS, register-resident sorted top-20 per
//           thread, then a shuffle-based parallel 16-way merge per query.
// Kernel B: gather neighbors, write [feature-center ; center], float4 stores.

#define NPTS   4096
#define NCH    64
#define NB     8
#define KSEL   20
#define QT     16      // queries per workgroup (one WMMA M tile)
#define CHUNK  512     // candidates staged per pass (16 x 512 f32 = 32KB LDS)
#define TPQ    16      // threads cooperating per query in selection

typedef __attribute__((ext_vector_type(2))) float v2f;
typedef __attribute__((ext_vector_type(8))) float v8f;

__device__ __forceinline__ bool better(float v1, int i1, float v2, int i2) {
    // total order: larger value first; ties -> smaller index first (top_k rule)
    return (v1 > v2) || (v1 == v2 && i1 < i2);
}

__global__ __launch_bounds__(256)
void knn_topk_kernel(const float* __restrict__ x, int* __restrict__ idx_out) {
    // 48KB static LDS: lsq[4096] (16KB) | ldist[16][512] (32KB)
    __shared__ __align__(16) char smem[49152];
    float* lsq   = (float*)smem;                 // NPTS floats
    float* ldist = (float*)(smem + 16384);       // QT*CHUNK floats

    const int b    = blockIdx.x / (NPTS / QT);
    const int q0   = (blockIdx.x % (NPTS / QT)) * QT;
    const int tid  = threadIdx.x;
    const int lane = tid & 31;
    const int wave = tid >> 5;            // 0..7
    const int laneLow = lane & 15;
    const bool hi  = lane >= 16;

    const float* xb = x + (size_t)b * NCH * NPTS;
    const float* X = xb;                  // channel 0
    const float* Y = xb + NPTS;           // channel 1
    const float* Z = xb + 2 * NPTS;       // channel 2

    // ---- precompute sq = |p|^2 into LDS ----
    for (int i = tid; i < NPTS; i += 256) {
        float xv = X[i], yv = Y[i], zv = Z[i];
        lsq[i] = xv * xv + yv * yv + zv * zv;
    }
    __syncthreads();

    // Per-lane base pointer so the hot-loop B-operand load is one
    // unconditional global_load (no saveexec in the loop).
    const float* pXZ = hi ? Z : X;

    // ---- A operand (queries), 32-bit A 16x4 layout:
    // lanes 0-15:  M=lane,    v0=K0(x), v1=K1(y)
    // lanes 16-31: M=lane-16, v0=K2(z), v1=K3(0)
    v2f aop;
    {
        int q = q0 + laneLow;
        aop.x = pXZ[q];
        aop.y = hi ? 0.0f : Y[q];
    }
    // per-lane query sq for the 8 C/D rows this lane holds (row M = r + 8*hi)
    float sqQ[8];
#pragma unroll
    for (int r = 0; r < 8; ++r) sqQ[r] = lsq[q0 + r + (hi ? 8 : 0)];

    // ---- per-thread register top-20 (descending), static indexing only ----
    float bv[KSEL];
    int   bi[KSEL];
#pragma unroll
    for (int j = 0; j < KSEL; ++j) { bv[j] = -INFINITY; bi[j] = 0x7fffffff; }

    const int g = tid >> 4;   // query row 0..15 owned in selection phase
    const int l = tid & 15;   // slot within query group

    for (int cb = 0; cb < NPTS; cb += CHUNK) {
        // ---- phase 1: WMMA distance tiles -> ldist (wave w owns 64 cols) ----
#pragma unroll
        for (int t = 0; t < CHUNK / (8 * 16); ++t) {
            const int cBase = cb + wave * (CHUNK / 8) + t * 16;
            const int c = cBase + laneLow;
            const float bx = pXZ[c];           // unconditional per-lane load
            const float by = Y[c];             // unconditional; masked below
            v2f bop;                           // B 4x16, mirrored layout over N
            bop.x = bx;
            bop.y = hi ? 0.0f : by;            // v_cndmask, no EXEC change
            v8f acc = {};
            acc = __builtin_amdgcn_wmma_f32_16x16x4_f32(
                /*neg_a=*/false, aop, /*neg_b=*/false, bop,
                /*c_mod=*/(short)0, acc, /*reuse_a=*/false, /*reuse_b=*/false);
            const float sqc = lsq[c];
            const int col = cBase - cb + laneLow;
#pragma unroll
            for (int r = 0; r < 8; ++r) {
                // C/D layout: lane holds row M = r + 8*hi, col N = laneLow
                float nd = 2.0f * acc[r] - sqQ[r] - sqc;
                ldist[(r + (hi ? 8 : 0)) * CHUNK + col] = nd;
            }
        }
        __syncthreads();

        // ---- phase 2: 16 threads/query scan chunk, update register top-20 ----
        for (int s = 0; s < CHUNK / TPQ; ++s) {
            const int cc = l + s * TPQ;
            const float v = ldist[g * CHUNK + cc];
            const int  ci = cb + cc;
            if (better(v, ci, bv[KSEL - 1], bi[KSEL - 1])) {
                bv[KSEL - 1] = v; bi[KSEL - 1] = ci;
#pragma unroll
                for (int j = KSEL - 1; j > 0; --j) {
                    if (!better(bv[j], bi[j], bv[j - 1], bi[j - 1])) break;
                    float tv = bv[j]; bv[j] = bv[j - 1]; bv[j - 1] = tv;
                    int   ti = bi[j]; bi[j] = bi[j - 1]; bi[j - 1] = ti;
                }
            }
        }
        __syncthreads();   // ldist reused next chunk
    }

    // ---- phase 3: parallel 16-way merge per query, all in registers ----
    // Each thread's bv/bi is sorted descending. 20 rounds: tournament-reduce
    // the 16 group heads via shfl_xor (width 16 == group), unique winner
    // (candidate indices are globally unique) pops its head by shifting.
    const int out_base = (b * NPTS + q0 + g) * KSEL;
    for (int sel = 0; sel < KSEL; ++sel) {
        float rv = bv[0];
        int   ri = bi[0];
#pragma unroll
        for (int off = 8; off >= 1; off >>= 1) {
            float ov = __shfl_xor(rv, off, TPQ);
            int   oi = __shfl_xor(ri, off, TPQ);
            if (better(ov, oi, rv, ri)) { rv = ov; ri = oi; }
        }
        if (l == 0) idx_out[out_base + sel] = ri;
        if (bi[0] == ri) {                 // this lane owned the winner: pop
#pragma unroll
            for (int j = 0; j < KSEL - 1; ++j) { bv[j] = bv[j + 1]; bi[j] = bi[j + 1]; }
            bv[KSEL - 1] = -INFINITY; bi[KSEL - 1] = 0x7fffffff;
        }
    }
}

__global__ __launch_bounds__(256)
void gather_feature_kernel(const float* __restrict__ x,
                           const int4* __restrict__ idx4,
                           float* __restrict__ out) {
    // one thread = (b, c, n, 4 neighbors); 8*64*4096*5 = 10,485,760 threads
    const unsigned t = blockIdx.x * 256u + threadIdx.x;
    const int kq = t % 5;
    unsigned r = t / 5;
    const int n = r % NPTS;  r /= NPTS;
    const int c = r % NCH;
    const int b = r / NCH;

    const int4 mi = idx4[(b * NPTS + n) * 5 + kq];
    const float* xc = x + ((size_t)b * NCH + c) * NPTS;
    const float center = xc[n];

    float4 f;
    f.x = xc[mi.x] - center;
    f.y = xc[mi.y] - center;
    f.z = xc[mi.z] - center;
    f.w = xc[mi.w] - center;

    float4* o1 = (float4*)(out + (((size_t)b * 128 + c) * NPTS + n) * KSEL + kq * 4);
    *o1 = f;
    float4 cv = make_float4(center, center, center, center);
    float4* o2 = (float4*)(out + (((size_t)b * 128 + 64 + c) * NPTS + n) * KSEL + kq * 4);
    *o2 = cv;
}

extern "C" void kernel_launch(void* const* d_in, const int* in_sizes, int n_in,
                              void* d_out, int out_size, void* d_ws, size_t ws_size,
                              hipStream_t stream) {
    const float* x = (const float*)d_in[0];
    float* out = (float*)d_out;
    int* idx = (int*)d_ws;   // NB*NPTS*KSEL ints = 2.62 MB scratch

    // Kernel A: 8 batches * 256 query-tiles of 16 = 2048 workgroups
    knn_topk_kernel<<<NB * (NPTS / QT), 256, 0, stream>>>(x, idx);

    // Kernel B: 10,485,760 threads / 256 = 40,960 blocks (exact)
    const unsigned totalB = (unsigned)NB * NCH * NPTS * (KSEL / 4);
    gather_feature_kernel<<<totalB / 256, 256, 0, stream>>>(x, (const int4*)idx, out);
}